// Quer_loss_prob_5class_90735479096049
// MI455X (gfx1250) — compile-verified
//
#include <hip/hip_runtime.h>
#include <hip/hip_bf16.h>
#include <math.h>

typedef __attribute__((ext_vector_type(2))) float v2f;
typedef __attribute__((ext_vector_type(8))) float v8f;

#define S_COMPACT 379
#define S_PAD     384
#define NTILES    24          // 384 / 16 columns per WMMA tile
#define EPSF      1e-6f

__device__ __forceinline__ float4 qmul_ref(const float4 p, const float4 q) {
    // reference _qua_mul, xyzw layout (w at index 3)
    float4 r;
    r.w = q.w*p.w - q.x*p.x - q.y*p.y - q.z*p.z;
    r.x = q.w*p.x + q.x*p.w + q.y*p.z - q.z*p.y;
    r.y = q.w*p.y - q.x*p.z + q.y*p.w + q.z*p.x;
    r.z = q.w*p.z + q.x*p.y - q.y*p.x + q.z*p.w;
    return r;
}

__device__ __forceinline__ void decode_cls(int c, int& cls, int& rem) {
    if      (c < 1)  { cls = 0; rem = c;      }
    else if (c < 3)  { cls = 1; rem = c - 1;  }
    else if (c < 7)  { cls = 2; rem = c - 3;  }
    else if (c < 19) { cls = 3; rem = c - 7;  }
    else             { cls = 4; rem = c - 19; }
}

extern "C" __global__ __launch_bounds__(256)
void quat_sym_loss_kernel(const float* __restrict__ x,
                          const float* __restrict__ y,
                          const int*   __restrict__ ncls,
                          const float* __restrict__ sym_qua,
                          float*       __restrict__ out,
                          int Btot)
{
    __shared__ float4 symLds[S_PAD];        // compacted table (epilogue qmul)
    __shared__ float2 bOp[NTILES * 32];     // operand-ready, conflict-free layout

    const int tid = threadIdx.x;

    // ---- stage compacted float4 table: offsets {0,1,3,7,19}, counts {1,2,4,12,360}
    for (int c = tid; c < S_PAD; c += blockDim.x) {
        float4 v = make_float4(0.f, 0.f, 0.f, 0.f);
        if (c < S_COMPACT) {
            int cls, rem; decode_cls(c, cls, rem);
            v = ((const float4*)sym_qua)[cls * 360 + rem];
        }
        symLds[c] = v;
    }
    // ---- stage B operands: entry (t, L) = the v2f lane L needs for tile t
    for (int e = tid; e < NTILES * 32; e += blockDim.x) {
        int t = e >> 5, L = e & 31;
        int col = t * 16 + (L & 15);
        float4 v = make_float4(0.f, 0.f, 0.f, 0.f);
        if (col < S_COMPACT) {
            int cls, rem; decode_cls(col, cls, rem);
            v = ((const float4*)sym_qua)[cls * 360 + rem];
        }
        bOp[e] = (L >> 4) ? make_float2(v.z, v.w)    // K=2,3 -> s.z, s.w
                          : make_float2(v.x, v.y);   // K=0,1 -> s.x, s.y
    }
    __syncthreads();

    const int lane = tid & 31;
    const int wave = tid >> 5;
    const int base = (blockIdx.x * (blockDim.x >> 5) + wave) * 16;  // 16 samples/wave
    const int half = lane >> 4;    // 0: D rows 0..7, A K=0..1 ; 1: D rows 8..15, A K=2..3
    const int lcol = lane & 15;    // column position within tile

    // ---- A operand: w-vector of sample base+lcol  (dots[b,s] = w(b) . s)
    int sA = base + lcol; if (sA >= Btot) sA = Btot - 1;
    float4 xv = ((const float4*)x)[sA];
    float4 yv = ((const float4*)y)[sA];
    float inv = rsqrtf(xv.x*xv.x + xv.y*xv.y + xv.z*xv.z + xv.w*xv.w);
    float px = xv.x*inv, py = xv.y*inv, pz = xv.z*inv, pw = xv.w*inv;
    float wx =  px*yv.w - py*yv.z + pz*yv.y - pw*yv.x;   // coeff of s.x
    float wy =  px*yv.z + py*yv.w - pz*yv.x - pw*yv.y;   // coeff of s.y
    float wz = -px*yv.y + py*yv.x + pz*yv.w - pw*yv.z;   // coeff of s.z
    float ww =  px*yv.x + py*yv.y + pz*yv.z + pw*yv.w;   // coeff of s.w
    v2f a;
    a[0] = half ? wz : wx;
    a[1] = half ? ww : wy;

    // ---- per-D-row class: rows r (0..7) map to samples base + half*8 + r
    int clsR[8];
#pragma unroll
    for (int r = 0; r < 8; ++r) {
        int s = base + half*8 + r; if (s >= Btot) s = Btot - 1;
        clsR[r] = ncls[s];
    }

    // ---- preload all B operands (conflict-free ds loads, single wait)
    v2f bArr[NTILES];
#pragma unroll
    for (int t = 0; t < NTILES; ++t) {
        float2 f = bOp[t * 32 + lane];
        v2f b; b[0] = f.x; b[1] = f.y;
        bArr[t] = b;
    }

    // ---- branchless lexicographic argmax: key = (|cc|bits << 32) | ~((sgn<<9)|col)
    unsigned long long best[8];
#pragma unroll
    for (int r = 0; r < 8; ++r) best[r] = 0ull;

    const float flo = -1.0f + EPSF, fhi = 1.0f - EPSF;

#pragma unroll
    for (int t = 0; t < NTILES; ++t) {
        int col = t * 16 + lcol;
        unsigned int colInv = ~(unsigned int)col;
        // exactly one class owns each column (ranges partition [0,379))
        int cls = (int)(col >= 1) + (int)(col >= 3) + (int)(col >= 7) + (int)(col >= 19);
        cls = (col < S_COMPACT) ? cls : 7;   // pad columns match no class

        v8f cz = {0.f,0.f,0.f,0.f,0.f,0.f,0.f,0.f};
        v8f d = __builtin_amdgcn_wmma_f32_16x16x4_f32(
                    false, a, false, bArr[t], (short)0, cz, false, false);
#pragma unroll
        for (int r = 0; r < 8; ++r) {
            // clip like reference; med3 == min(max(d,lo),hi), no canonicalize op
            float cc = __builtin_amdgcn_fmed3f(d[r], flo, fhi);
            unsigned int bits = __float_as_uint(cc);
            unsigned int khi  = bits & 0x7fffffffu;           // |cc| bits (monotone)
            unsigned int sgn9 = (bits >> 22) & 0x200u;        // sign -> branch bit 9
            unsigned int klo  = colInv ^ sgn9;                // = ~((sgn<<9)|col)
            unsigned long long key =
                ((unsigned long long)khi << 32) | (unsigned long long)klo;
            key = (cls == clsR[r]) ? key : 0ull;              // invalid -> loses
            best[r] = (key > best[r]) ? key : best[r];        // u64 max = lexicographic
        }
    }

    // ---- butterfly reduce across the 16 column-lanes of each half-wave
#pragma unroll
    for (int m = 1; m < 16; m <<= 1) {
#pragma unroll
        for (int r = 0; r < 8; ++r) {
            unsigned int bh = (unsigned int)(best[r] >> 32);
            unsigned int bl = (unsigned int)best[r];
            unsigned int oh = __shfl_xor(bh, m, 32);
            unsigned int ol = __shfl_xor(bl, m, 32);
            unsigned long long ok =
                ((unsigned long long)oh << 32) | (unsigned long long)ol;
            best[r] = (ok > best[r]) ? ok : best[r];
        }
    }

    // ---- epilogue: lanes 0..7 write rows 0..7, lanes 16..23 write rows 8..15
    int r = lane & 7;
    bool writer = (lcol < 8);
    int s = base + half*8 + r;
    if (writer && s < Btot) {
        unsigned int bh = (unsigned int)(best[r] >> 32);
        unsigned int meta = ~((unsigned int)best[r]);
        int col    = (int)(meta & 511u);
        int branch = (int)((meta >> 9) & 1u);
        float loss = acosf(__uint_as_float(bh));
        float4 xs = ((const float4*)x)[s];
        float4 ys = ((const float4*)y)[s];
        float4 sq = symLds[col];
        float4 pos = qmul_ref(ys, sq);                 // label_near = qmul(y, sym[j])
        float sg = branch ? -1.0f : 1.0f;              // x_near = idx<S ? x : -x

        out[s] = loss;                                  // loss [B,1]
        float* xo = out + Btot + (size_t)s * 4;         // x_near [B,4]
        xo[0] = sg*xs.x; xo[1] = sg*xs.y; xo[2] = sg*xs.z; xo[3] = sg*xs.w;
        float* ln = out + (size_t)Btot*5 + (size_t)s*4; // label_near [B,4]
        ln[0] = pos.x; ln[1] = pos.y; ln[2] = pos.z; ln[3] = pos.w;
    }
}

extern "C" void kernel_launch(void* const* d_in, const int* in_sizes, int n_in,
                              void* d_out, int out_size, void* d_ws, size_t ws_size,
                              hipStream_t stream) {
    const float* x   = (const float*)d_in[0];   // [B,4] f32
    const float* y   = (const float*)d_in[1];   // [B,4] f32
    // d_in[2] = pred_n, unused by forward
    const int*   n   = (const int*)d_in[3];     // [B] i32
    const float* sym = (const float*)d_in[4];   // [5,360,4] f32
    // d_in[5] = sym_mask, semantics hardcoded from SYM_COUNTS
    float* out = (float*)d_out;

    int B = in_sizes[0] / 4;
    int samplesPerBlock = 128;                  // 256 threads = 8 waves * 16 samples
    int grid = (B + samplesPerBlock - 1) / samplesPerBlock;
    quat_sym_loss_kernel<<<grid, 256, 0, stream>>>(x, y, n, sym, out, B);
}